// Hbv_76192719832027
// MI455X (gfx1250) — compile-verified
//
#include <hip/hip_runtime.h>
#include <math.h>
#include <stdint.h>

#define T_STEPS   730
#define LENF      15
#define NEARZERO  1e-5f
#define NPAR      14
#define BLK       64            // 2 wave32s per block
#define CHUNK     16            // timesteps staged per async chunk
#define ROW_FLTS  (BLK * 3)     // 192 floats = 768 bytes per staged timestep
#define ROW_BYTES (ROW_FLTS * 4)

__device__ __forceinline__ float sigmoidf_(float x) {
    return 1.0f / (1.0f + __expf(-x));
}

__global__ void __launch_bounds__(BLK)
hbv_uh_kernel(const float* __restrict__ x_phy,     // (T, G, 3)
              const float* __restrict__ params,    // (T, G, 14)
              float* __restrict__ out,             // (T, G)
              int T, int G)
{
    __shared__ float sbuf[2][CHUNK * ROW_FLTS];    // 24 KB double buffer

    const int tid  = threadIdx.x;
    const int wave = tid >> 5;
    const int lane = tid & 31;
    const int g    = blockIdx.x * BLK + tid;
    const int gc   = (g < G) ? g : (G - 1);        // clamp; no early return (barriers)
    const int g0   = blockIdx.x * BLK;             // first cell of this block

    // LDS byte offset of the staging buffer (low 32 bits of the generic address)
    const uint32_t ldsBase = (uint32_t)(uintptr_t)(&sbuf[0][0]);
    const uint32_t maxOff  = (uint32_t)((size_t)T * (size_t)G * 12u) - 8u;

    // ---------------- parameters (only last timestep is used) ----------------
    const float* pp = params + ((size_t)(T - 1) * (size_t)G + (size_t)gc) * NPAR;
    float s[NPAR];
    #pragma unroll
    for (int i = 0; i < NPAR; ++i) s[i] = sigmoidf_(pp[i]);

    const float BETA   = 1.0f   + s[0]  * 5.0f;
    const float FC     = 50.0f  + s[1]  * 950.0f;
    const float K0     = 0.05f  + s[2]  * 0.85f;
    const float K1     = 0.01f  + s[3]  * 0.49f;
    const float K2     = 0.001f + s[4]  * 0.199f;
    const float LP     = 0.2f   + s[5]  * 0.8f;
    const float PERCp  =          s[6]  * 10.0f;
    const float UZL    =          s[7]  * 100.0f;
    const float TT     = -2.5f  + s[8]  * 5.0f;
    const float CFMAX  = 0.5f   + s[9]  * 9.5f;
    const float CFR    =          s[10] * 0.1f;
    const float CWH    =          s[11] * 0.2f;
    const float rout_a =          s[12] * 2.9f;
    const float rout_b =          s[13] * 6.5f;

    const float invFC    = 1.0f / FC;
    const float invLPFC  = 1.0f / (LP * FC);
    const float CFRCFMAX = CFR * CFMAX;

    // -------- UH gamma weights; gamma-fn denominator cancels on normalize ----
    const float aa       = fmaxf(rout_a, 0.0f) + 0.1f;
    const float theta    = fmaxf(rout_b, 0.0f) + 0.5f;
    const float invTheta = 1.0f / theta;
    const float am1      = aa - 1.0f;

    float w[LENF];
    float wsum = 0.0f;
    #pragma unroll
    for (int k = 0; k < LENF; ++k) {
        const float tk = (float)k + 0.5f;
        const float u  = __powf(tk, am1) * __expf(-tk * invTheta);
        w[k] = u;
        wsum += u;
    }
    const float invWsum = 1.0f / wsum;
    #pragma unroll
    for (int k = 0; k < LENF; ++k) w[k] *= invWsum;

    // ---------------- state + convolution ring buffer (registers) -----------
    float SNOWPACK = 0.001f, MELTWATER = 0.001f, SM = 0.001f,
          SUZ = 0.001f, SLZ = 0.001f;
    float qh[LENF];
    #pragma unroll
    for (int k = 0; k < LENF; ++k) qh[k] = 0.0f;

    float* op = out + (size_t)gc;
    const int nChunks = (T + CHUNK - 1) / CHUNK;

    // -------- async DMA of one chunk: this wave stages 8 rows x 3 segs -------
    auto issue_chunk = [&](int c, int bufIdx) {
        const int t0 = c * CHUNK;
        #pragma unroll
        for (int rr = 0; rr < CHUNK / 2; ++rr) {
            const int r = wave * (CHUNK / 2) + rr;
            int t = t0 + r;
            t = (t < T) ? t : (T - 1);             // clamp remainder rows
            const uint32_t rowOff = (uint32_t)(((size_t)t * (size_t)G + (size_t)g0) * 12u);
            #pragma unroll
            for (int seg = 0; seg < 3; ++seg) {
                uint32_t off = rowOff + (uint32_t)seg * 256u + (uint32_t)lane * 8u;
                off = (off <= maxOff) ? off : maxOff;      // never overrun buffer
                const uint32_t ldsOff = ldsBase
                    + (uint32_t)bufIdx * (uint32_t)(CHUNK * ROW_BYTES)
                    + (uint32_t)r * (uint32_t)ROW_BYTES
                    + (uint32_t)seg * 256u + (uint32_t)lane * 8u;
                // GVS form: per-lane LDS dst, per-lane 32-bit offset, SGPR64 base
                asm volatile("global_load_async_to_lds_b64 %0, %1, %2"
                             :: "v"(ldsOff), "v"(off), "s"(x_phy)
                             : "memory");
            }
        }
    };

    issue_chunk(0, 0);

    for (int c = 0; c < nChunks; ++c) {
        const int buf = c & 1;
        if (c + 1 < nChunks) {
            issue_chunk(c + 1, (c + 1) & 1);
            // 24 ops of chunk c+1 in flight; in-order completion => <=24
            // outstanding means all of chunk c has landed in LDS.
            asm volatile("s_wait_asynccnt 0x18" ::: "memory");
        } else {
            asm volatile("s_wait_asynccnt 0x0" ::: "memory");
        }
        __syncthreads();                            // all waves' rows visible

        for (int r = 0; r < CHUNK; ++r) {
            const int t = c * CHUNK + r;
            if (t >= T) break;                      // uniform branch

            const float* rowp = &sbuf[buf][r * ROW_FLTS + tid * 3];
            const float Pt  = rowp[0];
            const float Tt  = rowp[1];
            const float PEt = rowp[2];

            // ----------------------- HBV step (reference order) --------------
            const float RAIN = (Tt >= TT) ? Pt : 0.0f;
            const float SNOW = (Tt <  TT) ? Pt : 0.0f;
            SNOWPACK += SNOW;
            const float melt = fminf(fmaxf(CFMAX * (Tt - TT), 0.0f), SNOWPACK);
            MELTWATER += melt;
            SNOWPACK  -= melt;
            const float refreeze = fminf(fmaxf(CFRCFMAX * (TT - Tt), 0.0f), MELTWATER);
            SNOWPACK  += refreeze;
            MELTWATER -= refreeze;
            const float tosoil = fmaxf(MELTWATER - CWH * SNOWPACK, 0.0f);
            MELTWATER -= tosoil;

            float sw = __powf(SM * invFC, BETA);
            sw = fminf(fmaxf(sw, 0.0f), 1.0f);
            const float recharge = (RAIN + tosoil) * sw;
            SM = SM + RAIN + tosoil - recharge;
            const float excess = fmaxf(SM - FC, 0.0f);
            SM -= excess;
            const float evapf = fminf(fmaxf(SM * invLPFC, 0.0f), 1.0f);
            const float ETact = fminf(PEt * evapf, SM);
            SM = fmaxf(SM - ETact, NEARZERO);

            SUZ = SUZ + recharge + excess;
            const float PERC = fminf(SUZ, PERCp);
            SUZ -= PERC;
            const float Q0 = K0 * fmaxf(SUZ - UZL, 0.0f);
            SUZ -= Q0;
            const float Q1 = K1 * SUZ;
            SUZ -= Q1;
            SLZ += PERC;
            const float Q2 = K2 * SLZ;
            SLZ -= Q2;
            const float Qsim = Q0 + Q1 + Q2;

            // --------------- fused UH convolution (register ring) ------------
            #pragma unroll
            for (int k = LENF - 1; k > 0; --k) qh[k] = qh[k - 1];
            qh[0] = Qsim;
            float flow = 0.0f;
            #pragma unroll
            for (int k = 0; k < LENF; ++k) flow = fmaf(w[k], qh[k], flow);

            if (g < G)
                __builtin_nontemporal_store(flow, op + (size_t)t * (size_t)G);
        }
        __syncthreads();       // safe to overwrite buf next iteration
    }
}

extern "C" void kernel_launch(void* const* d_in, const int* in_sizes, int n_in,
                              void* d_out, int out_size, void* d_ws, size_t ws_size,
                              hipStream_t stream) {
    const float* x_phy  = (const float*)d_in[0];   // (T, G, 3) float32
    const float* params = (const float*)d_in[1];   // (T, G, 14) float32
    float* out = (float*)d_out;                    // (T, G) float32

    const int T = T_STEPS;                         // fixed by reference
    const int G = out_size / T;                    // 10000

    const int grid = (G + BLK - 1) / BLK;          // 157 workgroups of 2 waves
    hbv_uh_kernel<<<grid, BLK, 0, stream>>>(x_phy, params, out, T, G);
}